// ResizerBackbone_85461259255934
// MI455X (gfx1250) — compile-verified
//
#include <hip/hip_runtime.h>
#include <hip/hip_bf16.h>

// Geometry fixed by the reference: B=16, C=512, T=4096, 5 levels.
// d_out layout (floats, return order = feats tuple then masks tuple):
//   feat0 @ 0          (16*512*4096)
//   feat1 @ 33554432   feat2 @ 50331648   feat3 @ 58720256   feat4 @ 62914560
//   mask0 @ 65011712   mask1 @ 65077248   mask2 @ 65110016
//   mask3 @ 65126400   mask4 @ 65134592   -> total 65138688
//
// d_ws layout: int L[16][5] @ 0 (320 B, padded to 512), uint8 mask0[16*4096] @ 512.

#if defined(__has_builtin)
#if __has_builtin(__builtin_amdgcn_tensor_load_to_lds) && \
    __has_builtin(__builtin_amdgcn_s_wait_tensorcnt)
#define USE_TDM 1
#endif
#endif

typedef unsigned int u32x4 __attribute__((ext_vector_type(4)));
typedef int i32x8 __attribute__((ext_vector_type(8)));
typedef int i32x4 __attribute__((ext_vector_type(4)));

// ---------------------------------------------------------------------------
// Kernel 1: mask pyramid + valid-length table + byte-mask stash.
// Scale factor is exactly 2 -> interp weight is exactly 0.5 ->
// mask_out[j] = mask_in[2j] && mask_in[2j+1] (bit-exact vs "0.5*(a+b) > 0.5").
// One block per batch; Lws[b*5+lev] = count of valid (False) entries.
// ---------------------------------------------------------------------------
__global__ __launch_bounds__(256) void mask_levels_kernel(
    const int* __restrict__ mask, float* __restrict__ out,
    int* __restrict__ Lws, unsigned char* __restrict__ mws) {
  __shared__ unsigned char mcur[4096];
  __shared__ unsigned char mnxt[2048];
  __shared__ int cnt;
  const long long MOFF[5] = {65011712LL, 65077248LL, 65110016LL, 65126400LL, 65134592LL};

  const int b = blockIdx.x;
  const int tid = threadIdx.x;

  if (tid == 0) cnt = 0;
  __syncthreads();

  int local = 0;
  for (int t = tid; t < 4096; t += 256) {
    const int mv = (mask[(size_t)b * 4096 + t] != 0) ? 1 : 0;
    mcur[t] = (unsigned char)mv;
    mws[(size_t)b * 4096 + t] = (unsigned char)mv;          // byte mask for fused kernel
    out[MOFF[0] + (size_t)b * 4096 + t] = (float)mv;        // mask level 0 output
    local += (mv == 0);
  }
  atomicAdd(&cnt, local);
  __syncthreads();
  if (tid == 0) Lws[b * 5 + 0] = cnt;

  int Tl = 4096;
  for (int lev = 1; lev <= 4; ++lev) {
    Tl >>= 1;
    __syncthreads();
    if (tid == 0) cnt = 0;
    __syncthreads();
    int lc = 0;
    for (int j = tid; j < Tl; j += 256) {
      const int mv = (int)(mcur[2 * j] & mcur[2 * j + 1]);
      mnxt[j] = (unsigned char)mv;
      out[MOFF[lev] + (size_t)b * Tl + j] = (float)mv;
      lc += (mv == 0);
    }
    atomicAdd(&cnt, lc);
    __syncthreads();
    if (tid == 0) Lws[b * 5 + lev] = cnt;
    for (int j = tid; j < Tl; j += 256) mcur[j] = mnxt[j];
  }
}

// ---------------------------------------------------------------------------
// Kernel 2: fused feature pyramid. One block per (b, c) row.
// Phase 1: TDM DMA of the 16 KB source row into LDS (wave 0 issues one
//          tensor_load_to_lds, s_wait_tensorcnt 0, then workgroup barrier).
// Phase 2: mask pass -> feat0 (global) + masked row back into LDS.
// Phase 3: levels 1..4 computed entirely from LDS (ping-pong 16KB/8KB),
//          each level written to global once. Zero intermediate global reads.
// ---------------------------------------------------------------------------
__global__ __launch_bounds__(256) void pyramid_kernel(
    const float* __restrict__ x, const int* __restrict__ Lws,
    const unsigned char* __restrict__ mws, float* __restrict__ out) {
  __shared__ float bufA[4096];
  __shared__ float bufB[2048];

  const int tid = threadIdx.x;
  const int c = blockIdx.x;          // 0..511
  const int b = blockIdx.y;          // 0..15
  const int row = b * 512 + c;
  const size_t FOFF[5] = {0ull, 33554432ull, 50331648ull, 58720256ull, 62914560ull};

  const float* xrow = x + (size_t)row * 4096;

  // ---- Phase 1: stage raw source row into LDS ----
#ifdef USE_TDM
  if (tid < 32) {  // one wave issues the DMA; TDM ignores EXEC
    const unsigned long long ga = (unsigned long long)(uintptr_t)xrow;
    const unsigned lds_addr = (unsigned)(uintptr_t)(&bufA[0]);  // generic[31:0] = LDS byte offset
    // D# group0: count=1, lds_addr, global_addr[56:0], type=2 ("image")
    u32x4 g0;
    g0.x = 1u;
    g0.y = lds_addr;
    g0.z = (unsigned)(ga & 0xFFFFFFFFull);
    g0.w = (unsigned)((ga >> 32) & 0x01FFFFFFull) | (2u << 30);
    // D# group1 (bit-packed, see ISA 8.4):
    //  data_size=2 (4B) @ [17:16]; tensor_dim0=4096 @ [79:48]; tensor_dim1=1 @ [111:80];
    //  tile_dim0=4096 @ [127:112]; tile_dim1=1 @ [143:128]; tensor_dim0_stride=4096 @ [207:160]
    const i32x8 g1 = {0x00020000, 0x10000000, 0x00010000, 0x10000000,
                      0x00000001, 0x00001000, 0x00000000, 0x00000000};
    const i32x4 z4 = {0, 0, 0, 0};
    const i32x8 z8 = {0, 0, 0, 0, 0, 0, 0, 0};
    // clang-23 lane: 6-arg form (g0, g1, g2, g3, extra-group, cpol)
    __builtin_amdgcn_tensor_load_to_lds(g0, g1, z4, z4, z8, 0);
    __builtin_amdgcn_s_wait_tensorcnt(0);
  }
#else
  for (int k = 0; k < 4; ++k) {
    const int i4 = tid + 256 * k;  // float4 index within row (1024 total)
    const float4 v = reinterpret_cast<const float4*>(xrow)[i4];
    *reinterpret_cast<float4*>(&bufA[4 * i4]) = v;
  }
#endif
  __syncthreads();

  // ---- Phase 2: mask pass -> feat0 + masked LDS row ----
  const unsigned char* mrow = mws + (size_t)b * 4096;
  float* orow0 = out + (size_t)row * 4096;  // feat0 offset 0
  for (int k = 0; k < 4; ++k) {
    const int i4 = tid + 256 * k;
    float4 v = *reinterpret_cast<float4*>(&bufA[4 * i4]);
    const uchar4 m = *reinterpret_cast<const uchar4*>(&mrow[4 * i4]);
    v.x = m.x ? 0.0f : v.x;
    v.y = m.y ? 0.0f : v.y;
    v.z = m.z ? 0.0f : v.z;
    v.w = m.w ? 0.0f : v.w;
    *reinterpret_cast<float4*>(&bufA[4 * i4]) = v;
    reinterpret_cast<float4*>(orow0)[i4] = v;
  }
  __syncthreads();

  // ---- Phase 3: levels 1..4 from LDS ----
  float* cur = bufA;
  float* nxt = bufB;
  int Tout = 4096;
  for (int lev = 1; lev <= 4; ++lev) {
    Tout >>= 1;
    const int Lin = Lws[b * 5 + lev - 1];
    const int Lout = Lws[b * 5 + lev];
    const int LoutS = (Lout > 0) ? Lout : 1;
    const float scale = (float)Lin / (float)LoutS;   // exact: fp32 divide as reference
    const int hi = (Lin > 0) ? (Lin - 1) : 0;
    float* o = out + FOFF[lev] + (size_t)row * (size_t)Tout;

    for (int j = tid; j < Tout; j += 256) {
      float v = 0.0f;
      if (j < Lout) {
        // mul-then-sub kept un-fused to match reference floor() boundaries
        const float s = fmaxf(__fsub_rn(__fmul_rn((float)j + 0.5f, scale), 0.5f), 0.0f);
        int k0 = (int)floorf(s);
        if (k0 > hi) k0 = hi;
        const int k1 = (k0 + 1 > hi) ? hi : (k0 + 1);
        const float w = __fsub_rn(s, (float)k0);     // weight uses CLAMPED k0
        const float w1 = __fsub_rn(1.0f, w);
        v = __fadd_rn(__fmul_rn(cur[k0], w1), __fmul_rn(cur[k1], w));
      }
      if (lev < 4) nxt[j] = v;
      o[j] = v;   // j >= Lout rows must be zero (d_out is poisoned)
    }
    __syncthreads();
    float* t = cur; cur = nxt; nxt = t;
  }
}

// ---------------------------------------------------------------------------
extern "C" void kernel_launch(void* const* d_in, const int* in_sizes, int n_in,
                              void* d_out, int out_size, void* d_ws, size_t ws_size,
                              hipStream_t stream) {
  const float* x  = (const float*)d_in[0];  // [16, 512, 4096] f32
  const int* mask = (const int*)d_in[1];    // [16, 4096] bool -> int32
  float* out = (float*)d_out;
  int* Lws = (int*)d_ws;                                     // [16][5] lengths
  unsigned char* mws = (unsigned char*)d_ws + 512;           // [16][4096] byte mask

  (void)in_sizes; (void)n_in; (void)out_size; (void)ws_size;

  // 1) mask pyramid + length table + byte-mask stash (tiny)
  mask_levels_kernel<<<dim3(16), dim3(256), 0, stream>>>(mask, out, Lws, mws);

  // 2) fused feature pyramid: TDM row staging + all 4 resize levels from LDS
  pyramid_kernel<<<dim3(512, 16), dim3(256), 0, stream>>>(x, Lws, mws, out);
}